// distregression_36326833390061
// MI455X (gfx1250) — compile-verified
//
#include <hip/hip_runtime.h>
#include <hip/hip_bf16.h>

// distregression: per-pixel split-variance argmin over n=10 planes.
// Memory-bound streaming kernel (22 MB @ 23.3 TB/s ~ 1 us); uses the CDNA5
// async global->LDS data mover path for deep prefetch + VGPR-lean two-pass
// compute out of LDS.

#define AS3 __attribute__((address_space(3)))

namespace {
constexpr int N_PLANES = 10;
constexpr int HW       = 256 * 256;     // H*W, power of two (shift 16)
constexpr int TPB      = 256;           // 8 waves of 32
constexpr int PPT      = 4;             // pixels per thread (float4 / b128)
constexpr int PPB      = TPB * PPT;     // 1024 pixels per block
constexpr int LDS_FLOATS = 2 * N_PLANES * PPB;  // 20480 floats = 80 KB
}

__global__ __launch_bounds__(TPB) void distreg_kernel(
    const float* __restrict__ s1, const float* __restrict__ blur,
    float* __restrict__ out)
{
    __shared__ __align__(16) float smem[LDS_FLOATS];
    // layout: plane j -> s1 tile at [j*2*PPB, +PPB), blur tile at [+PPB, +2*PPB)

    const unsigned tid   = threadIdx.x;
    const unsigned p0    = blockIdx.x * PPB;        // first pixel of this block
    const unsigned b     = p0 >> 16;                // p0 / HW  (HW = 2^16)
    const unsigned r     = (p0 & (HW - 1)) + tid * PPT;
    const unsigned gbyte = (b * (unsigned)(N_PLANES * HW) + r) * 4u;

    // Raw LDS offset for the async-to-LDS instructions (addrspace(3) value).
    const unsigned lds_base = (unsigned)(unsigned long long)(AS3 float*)smem;
    const unsigned long long s1a = (unsigned long long)s1;
    const unsigned long long bla = (unsigned long long)blur;

    // ---- Deep async prefetch: 20 outstanding b128 transfers per lane. ----
    // Each lane stages its own 16B slice per plane; it only ever reads back
    // the LDS it wrote itself, so no workgroup barrier is required.
#pragma unroll
    for (int j = 0; j < N_PLANES; ++j) {
        unsigned l_s1  = lds_base + (unsigned)(j * (2 * PPB)) * 4u + tid * 16u;
        unsigned l_bl  = l_s1 + (unsigned)PPB * 4u;
        unsigned g_off = gbyte + (unsigned)j * (unsigned)(HW * 4);
        asm volatile("global_load_async_to_lds_b128 %0, %1, %2"
                     :: "v"(l_s1), "v"(g_off), "s"(s1a) : "memory");
        asm volatile("global_load_async_to_lds_b128 %0, %1, %2"
                     :: "v"(l_bl), "v"(g_off), "s"(bla) : "memory");
    }
    asm volatile("s_wait_asynccnt 0" ::: "memory");

    // ---- Pass 1: S0 = sum(lo), Q0 = sum(lo^2) over planes (split i=0). ----
    float S0[PPT] = {0.f, 0.f, 0.f, 0.f};
    float Q0[PPT] = {0.f, 0.f, 0.f, 0.f};
#pragma unroll
    for (int j = 0; j < N_PLANES; ++j) {
        const float4 a = *(const float4*)&smem[j * (2 * PPB) + tid * PPT];
        const float4 u = *(const float4*)&smem[j * (2 * PPB) + PPB + tid * PPT];
        const float av[PPT] = {a.x, a.y, a.z, a.w};
        const float uv[PPT] = {u.x, u.y, u.z, u.w};
#pragma unroll
        for (int k = 0; k < PPT; ++k) {
            const float lo = av[k] - uv[k];
            S0[k] += lo;
            Q0[k] = fmaf(lo, lo, Q0[k]);
        }
    }

    // ---- Pass 2: incremental split sweep + argmin of n*Q - S^2. ----
    // split i swaps plane i-1 lo->hi:  S += 2*blur,  Q += 4*s1*blur.
    // std_i ordering == ordering of f_i = n*Q_i - S_i^2 (monotone).
    float S[PPT], Q[PPT], fbest[PPT], Sbest[PPT];
#pragma unroll
    for (int k = 0; k < PPT; ++k) {
        S[k] = S0[k];
        Q[k] = Q0[k];
        fbest[k] = fmaf((float)N_PLANES, Q0[k], -S0[k] * S0[k]);
        Sbest[k] = S0[k];
    }
#pragma unroll
    for (int i = 1; i <= N_PLANES; ++i) {
        const int j = i - 1;
        const float4 a = *(const float4*)&smem[j * (2 * PPB) + tid * PPT];
        const float4 u = *(const float4*)&smem[j * (2 * PPB) + PPB + tid * PPT];
        const float av[PPT] = {a.x, a.y, a.z, a.w};
        const float uv[PPT] = {u.x, u.y, u.z, u.w};
#pragma unroll
        for (int k = 0; k < PPT; ++k) {
            S[k] = fmaf(2.0f, uv[k], S[k]);
            Q[k] = fmaf(4.0f * av[k], uv[k], Q[k]);
            const float f = fmaf((float)N_PLANES, Q[k], -S[k] * S[k]);
            if (f < fbest[k]) {   // strict < : first minimum wins (argmin semantics)
                fbest[k] = f;
                Sbest[k] = S[k];
            }
        }
    }

    // Output = mean of winning candidate = S_best / n.
    float4 o;
    o.x = Sbest[0] * (1.0f / N_PLANES);
    o.y = Sbest[1] * (1.0f / N_PLANES);
    o.z = Sbest[2] * (1.0f / N_PLANES);
    o.w = Sbest[3] * (1.0f / N_PLANES);
    *(float4*)&out[p0 + tid * PPT] = o;
}

extern "C" void kernel_launch(void* const* d_in, const int* in_sizes, int n_in,
                              void* d_out, int out_size, void* d_ws, size_t ws_size,
                              hipStream_t stream) {
    (void)n_in; (void)d_ws; (void)ws_size; (void)out_size;
    const float* s1   = (const float*)d_in[0];
    const float* blur = (const float*)d_in[1];
    float* out = (float*)d_out;

    const int npix = in_sizes[0] / N_PLANES;   // bs*H*W = 262144
    const int grid = npix / PPB;               // 256 blocks
    distreg_kernel<<<grid, TPB, 0, stream>>>(s1, blur, out);
}